// GraphConv_39917426049651
// MI455X (gfx1250) — compile-verified
//
#include <hip/hip_runtime.h>

typedef __attribute__((ext_vector_type(2))) float v2f;
typedef __attribute__((ext_vector_type(8))) float v8f;

#define D 128            // D_IN == D_OUT == 128
#define TN 8             // 128/16 N-tiles

// ---------------------------------------------------------------------------
// Kernel 1: zero the output accumulator (d_out is poisoned by the harness).
// ---------------------------------------------------------------------------
__global__ void zero_f4(float4* __restrict__ p, long n4) {
    long i = (long)blockIdx.x * blockDim.x + threadIdx.x;
    long stride = (long)gridDim.x * blockDim.x;
    for (; i < n4; i += stride) p[i] = make_float4(0.f, 0.f, 0.f, 0.f);
}

// ---------------------------------------------------------------------------
// Kernel 2: support = x @ W with V_WMMA_F32_16X16X4_F32.
// One wave per 16x16 output tile; blockIdx.x = M-tile, wave id = N-tile.
// ---------------------------------------------------------------------------
__global__ __launch_bounds__(256)
void gemm_support(const float* __restrict__ x, const float* __restrict__ w,
                  float* __restrict__ support, int n_nodes) {
    const int lane  = threadIdx.x & 31;
    const int tileM = blockIdx.x;           // 16 rows of x
    const int tileN = threadIdx.x >> 5;     // 16 cols of W (8 waves = full N)

    const int m  = lane & 15;               // A: row within tile, B/C: column
    const int kb = (lane >> 4) << 1;        // K sub-offset: 0 (lanes 0-15) or 2

    int row = tileM * 16 + m;
    if (row >= n_nodes) row = n_nodes - 1;  // clamp loads (uniform for N=100000)

    const float* arow = x + (size_t)row * D + kb;          // A[m][kb + k0 + {0,1}]
    const float* bcol = w + (size_t)kb * D + tileN * 16 + m; // B[kb + k0 + {0,1}][n]

    v8f c = {};
#pragma unroll
    for (int k0 = 0; k0 < D; k0 += 4) {
        v2f a, b;
        a.x = arow[k0];
        a.y = arow[k0 + 1];
        b.x = bcol[(size_t)k0 * D];
        b.y = bcol[(size_t)(k0 + 1) * D];
        // D = A(16x4) * B(4x16) + C  -> v_wmma_f32_16x16x4_f32
        c = __builtin_amdgcn_wmma_f32_16x16x4_f32(
                /*neg_a=*/false, a, /*neg_b=*/false, b,
                /*c_mod=*/(short)0, c, /*reuse_a=*/false, /*reuse_b=*/false);
    }

    // C/D layout: VGPR r -> row (r + 0|8), col = lane&15
    const int rowoff = (lane >> 4) << 3;    // 0 or 8
#pragma unroll
    for (int r = 0; r < 8; ++r) {
        int orow = tileM * 16 + rowoff + r;
        if (orow < n_nodes)
            support[(size_t)orow * D + tileN * 16 + m] = c[r];
    }
}

// ---------------------------------------------------------------------------
// Kernel 3: edge scatter. One wave per edge, one float4 per lane (32*16B =
// one 512B row). Gathers are L2-resident (support = 51MB << 192MB L2);
// prefetch the next edge's source row to hide L2 latency.
// ---------------------------------------------------------------------------
__global__ __launch_bounds__(256)
void spmm_edges(const int* __restrict__ rows, const int* __restrict__ cols,
                const float* __restrict__ vals, const float* __restrict__ support,
                float* __restrict__ out, int E) {
    const int lane = threadIdx.x & 31;
    int e       = (int)(((long)blockIdx.x * blockDim.x + threadIdx.x) >> 5);
    const int n = (int)(((long)gridDim.x * blockDim.x) >> 5);

    for (; e < E; e += n) {
        const int   r = rows[e];
        const int   c = cols[e];
        const float v = vals[e];

        int enext = e + n;
        if (enext < E) {
            int cn = cols[enext];
            __builtin_prefetch(support + (size_t)cn * D + lane * 4, 0, 0);
        }

        const float4 g = ((const float4*)(support + (size_t)c * D))[lane];
        float* dst = out + (size_t)r * D + lane * 4;
        atomicAdd(dst + 0, g.x * v);
        atomicAdd(dst + 1, g.y * v);
        atomicAdd(dst + 2, g.z * v);
        atomicAdd(dst + 3, g.w * v);
    }
}

// ---------------------------------------------------------------------------
// Kernel 4: out = out / norm[row] + bias[col], vectorized float4.
// ---------------------------------------------------------------------------
__global__ void finalize(float* __restrict__ out, const float* __restrict__ norm,
                         const float* __restrict__ bias, long n4) {
    long i = (long)blockIdx.x * blockDim.x + threadIdx.x;
    if (i >= n4) return;
    const int  row = (int)(i >> 5);         // 32 float4 per row of 128
    const int  c4  = (int)(i & 31);
    const float inv = 1.0f / norm[row];
    float4 v = ((float4*)out)[i];
    const float4 b = ((const float4*)bias)[c4];
    v.x = v.x * inv + b.x;
    v.y = v.y * inv + b.y;
    v.z = v.z * inv + b.z;
    v.w = v.w * inv + b.w;
    ((float4*)out)[i] = v;
}

// ---------------------------------------------------------------------------
extern "C" void kernel_launch(void* const* d_in, const int* in_sizes, int n_in,
                              void* d_out, int out_size, void* d_ws, size_t ws_size,
                              hipStream_t stream) {
    const float* x      = (const float*)d_in[0];   // [N, 128]
    const float* weight = (const float*)d_in[1];   // [128, 128]
    const float* bias   = (const float*)d_in[2];   // [128]
    const int*   rows   = (const int*)  d_in[3];   // [E]
    const int*   cols   = (const int*)  d_in[4];   // [E]
    const float* vals   = (const float*)d_in[5];   // [E]
    const float* norm   = (const float*)d_in[6];   // [N, 1]

    const int n_nodes = in_sizes[0] / D;           // 100000
    const int E       = in_sizes[3];               // 3200000

    float* support = (float*)d_ws;                 // [N, 128] fp32 scratch (51.2 MB)
    float* out     = (float*)d_out;                // [N, 128] fp32

    const long n4 = (long)n_nodes * (D / 4);       // float4 elements of out

    // 1) zero the accumulator
    zero_f4<<<1024, 256, 0, stream>>>((float4*)out, n4);

    // 2) support = x @ W  (WMMA f32 16x16x4)
    const int mtiles = (n_nodes + 15) / 16;        // 6250
    gemm_support<<<mtiles, 256, 0, stream>>>(x, weight, support, n_nodes);

    // 3) edge gather/scale/scatter-add
    spmm_edges<<<4096, 256, 0, stream>>>(rows, cols, vals, support, out, E);

    // 4) out = out / norm + bias
    finalize<<<(unsigned)((n4 + 255) / 256), 256, 0, stream>>>(out, norm, bias, n4);
}